// VADetector_44358422233743
// MI455X (gfx1250) — compile-verified
//
#include <hip/hip_runtime.h>
#include <stdint.h>

// ---------------------------------------------------------------------------
// Viterbi detector, MI455X (gfx1250, wave32).
//
// T = 2^20 steps, 16 nominal states. Since table[s]==table[s+8] and the
// initial metric vector is zero, the metric vector is 8-periodic: we carry
// only u[0..7]. Decode is block-parallel: 16384 blocks of L=64 steps, each
// preceded by a W=32 step warm-up from zero metrics (survivor-merge window,
// >= 8x channel memory). One block per LANE: the entire ACS recurrence lives
// in VGPRs; there is no cross-lane traffic in the hot loop.
//
// Data movement: each lane's chunk of y is a contiguous 32-float run, runs
// strided L floats across lanes -> a 2D tile. One TENSOR_LOAD_TO_LDS (TDM)
// per workgroup per chunk stages it, using TDM LDS padding (32 dwords data +
// 4 dwords pad -> 36-dword row pitch) to de-conflict the per-lane
// ds_load_b128 reads. Bits are written from registers with b128 stores
// (4 MB total; L2 absorbs the per-lane striding).
// ---------------------------------------------------------------------------

#define T_LEN   (1u << 20)
#define NST     16
#define NS8     8
#define L_BLK   64                  // decoded steps per block
#define W_WARM  32                  // warm-up steps (== CHUNK)
#define CHUNK   32                  // steps staged per TDM load
#define NCH     (L_BLK / CHUNK)     // 2 decoded chunks per block
#define BPG     128                 // blocks per workgroup == threads per WG
#define NBLK    (T_LEN / L_BLK)     // 16384 blocks
#define NWG     (NBLK / BPG)        // 128 workgroups
#define ROWP    36                  // LDS row pitch in dwords (32 data + 4 pad)

typedef uint32_t u32x4 __attribute__((ext_vector_type(4)));
typedef int      i32x4 __attribute__((ext_vector_type(4)));
typedef int      i32x8 __attribute__((ext_vector_type(8)));

// Set bits [lo, lo+len) of a little-endian packed descriptor.
static __device__ __forceinline__ void dset(uint32_t* w, int lo, int len, uint64_t v) {
    int word = lo >> 5, bit = lo & 31;
    uint64_t m = (len >= 64) ? ~0ull : ((1ull << len) - 1ull);
    v &= m;
    w[word] |= (uint32_t)(v << bit);
    if (bit + len > 32) w[word + 1] |= (uint32_t)(v >> (32 - bit));
}

// Issue one 2D TDM load: tile_rows rows of CHUNK f32 elements, rows strided
// L_BLK floats in memory, landing in LDS with 4-dword padding per 32-dword
// row (pad_interval code 4 -> 32 dwords, pad_amount code 3 -> 4 dwords).
static __device__ __forceinline__ void tdm_load_2d(uint32_t lds_byte_addr,
                                                   const void* gaddr,
                                                   uint32_t tile_rows) {
    uint32_t g0[4] = {0, 0, 0, 0};
    uint32_t g1[8] = {0, 0, 0, 0, 0, 0, 0, 0};
    uint32_t g2[4] = {0, 0, 0, 0};
    uint32_t g3[4] = {0, 0, 0, 0};
    uint64_t ga = (uint64_t)(uintptr_t)gaddr;

    // Group 0
    dset(g0, 0, 2, 1);                    // count = 1 valid descriptor
    dset(g0, 32, 32, lds_byte_addr);      // lds_addr (bytes)
    dset(g0, 64, 57, ga);                 // global_addr (tile start)
    dset(g0, 126, 2, 2);                  // type = 2 ("image")

    // Group 1
    dset(g1, 16, 2, 2);                   // data_size = 2 -> 4 bytes
    dset(g1, 20, 1, 1);                   // pad_enable
    dset(g1, 22, 3, 4);                   // pad_interval: code 4 -> 32 dwords
    dset(g1, 25, 7, 3);                   // pad_amount:   code 3 -> 4 dwords
    dset(g1, 48, 32, CHUNK);              // tensor_dim0
    dset(g1, 80, 32, 0x7FFFFFFFu);        // tensor_dim1 (no OOB clipping)
    dset(g1, 112, 16, CHUNK);             // tile_dim0 = 32 elements
    dset(g1, 128, 16, tile_rows);         // tile_dim1 = rows (tile_dim2 = 0)
    dset(g1, 160, 48, L_BLK);             // tensor_dim0_stride = L floats

    // Groups 2/3: benign 1-length upper dims, tile_dim3/4 = 0 (unused).
    dset(g2, 0, 32, 1);                   // tensor_dim2 = 1
    dset(g2, 32, 32, 1);                  // tensor_dim3 = 1
    dset(g3, 48, 32, 1);                  // tensor_dim4 = 1

    u32x4 v0 = {g0[0], g0[1], g0[2], g0[3]};
    i32x8 v1 = {(int)g1[0], (int)g1[1], (int)g1[2], (int)g1[3],
                (int)g1[4], (int)g1[5], (int)g1[6], (int)g1[7]};
    i32x4 v2 = {(int)g2[0], (int)g2[1], (int)g2[2], (int)g2[3]};
    i32x4 v3 = {(int)g3[0], (int)g3[1], (int)g3[2], (int)g3[3]};
    i32x8 v4 = {0, 0, 0, 0, 0, 0, 0, 0};  // extra operand in 6-arg builtin
    __builtin_amdgcn_tensor_load_to_lds(v0, v1, v2, v3, v4, 0);
}

// One ACS step on the 8-state reduced trellis (16 branch metrics).
static __device__ __forceinline__ void acs_step(float u[NS8], const float p[NST],
                                                float yv) {
    float m[NST];
#pragma unroll
    for (int j = 0; j < NST; ++j)
        m[j] = u[j & 7] + __builtin_fabsf(yv - p[j]);   // abs folds as src mod
    float nu[NS8];
#pragma unroll
    for (int s = 0; s < NS8; ++s)
        nu[s] = fminf(m[2 * s], m[2 * s + 1]);
#pragma unroll
    for (int s = 0; s < NS8; ++s) u[s] = nu[s];
}

// Parity of first-occurrence argmin over u[0..7] (matches jnp.argmin ties).
static __device__ __forceinline__ float argmin_parity(const float u[NS8]) {
    float bv = u[0];
    int bi = 0;
#pragma unroll
    for (int i = 1; i < NS8; ++i) {
        bool lt = u[i] < bv;       // strict < keeps earliest index on ties
        bv = lt ? u[i] : bv;
        bi = lt ? i : bi;
    }
    return (float)(bi & 1);
}

__global__ __launch_bounds__(BPG) void VADetector_44358422233743_kernel(
    const float* __restrict__ y, const float* __restrict__ sp,
    float* __restrict__ out) {
    __shared__ __align__(16) float ybuf[BPG * ROWP];

    const int tid = (int)threadIdx.x;
    const int wg = (int)blockIdx.x;
    const int firstBlk = wg * BPG;
    const int gblk = firstBlk + tid;
    const int t0 = gblk * L_BLK;

    float p[NST];
#pragma unroll
    for (int j = 0; j < NST; ++j) p[j] = sp[j];

    float u[NS8];
#pragma unroll
    for (int s = 0; s < NS8; ++s) u[s] = 0.0f;

    // Chunks: c = -1 is the warm-up window [t0-W, t0); c = 0..NCH-1 decode.
    for (int c = -1; c < NCH; ++c) {
        __syncthreads();  // previous chunk fully consumed before LDS reuse
        if (tid < 32) {   // wave 0 issues the TDM op (wave-uniform branch)
            long base_t = (long)firstBlk * L_BLK + (long)c * CHUNK;
            uint32_t lds = (uint32_t)(uintptr_t)&ybuf[0];
            uint32_t rows = BPG;
            const float* ga = y + base_t;
            if (c < 0 && wg == 0) {
                // Block 0 has no warm-up (true start is zeros at t=0); shift
                // the tile to block 1's window to avoid reads before y[0].
                ga += L_BLK;
                lds += ROWP * 4;
                rows = BPG - 1;
            }
            tdm_load_2d(lds, ga, rows);
            __builtin_amdgcn_s_wait_tensorcnt(0);
        }
        __syncthreads();  // TDM-written LDS visible to all waves

        if (c < 0) {
            if (gblk != 0) {  // block 0 keeps u = 0 through the warm-up
#pragma unroll
                for (int k4 = 0; k4 < CHUNK / 4; ++k4) {
                    float4 yq = *(const float4*)&ybuf[tid * ROWP + k4 * 4];
                    acs_step(u, p, yq.x);
                    acs_step(u, p, yq.y);
                    acs_step(u, p, yq.z);
                    acs_step(u, p, yq.w);
                }
            }
        } else {
#pragma unroll
            for (int k4 = 0; k4 < CHUNK / 4; ++k4) {
                float4 yq = *(const float4*)&ybuf[tid * ROWP + k4 * 4];
                float yv[4] = {yq.x, yq.y, yq.z, yq.w};
                float b[4];
#pragma unroll
                for (int q = 0; q < 4; ++q) {
                    b[q] = argmin_parity(u);  // bit emitted BEFORE the update
                    acs_step(u, p, yv[q]);
                }
                float4 bq = {b[0], b[1], b[2], b[3]};
                *(float4*)&out[t0 + c * CHUNK + k4 * 4] = bq;
            }
        }
    }
}

extern "C" void kernel_launch(void* const* d_in, const int* in_sizes, int n_in,
                              void* d_out, int out_size, void* d_ws,
                              size_t ws_size, hipStream_t stream) {
    (void)in_sizes; (void)n_in; (void)out_size; (void)d_ws; (void)ws_size;
    const float* y = (const float*)d_in[0];         // [1, T] float32
    const float* sp = (const float*)d_in[1];        // [16, 1] float32 priors
    float* out = (float*)d_out;                     // [T] float32 bits
    hipLaunchKernelGGL(VADetector_44358422233743_kernel, dim3(NWG), dim3(BPG),
                       0, stream, y, sp, out);
}